// MMSolver_5738076307875
// MI455X (gfx1250) — compile-verified
//
#include <hip/hip_runtime.h>
#include <cstddef>

#define N 128
#define NN (N * N)
#define NB 16
#define T_STEPS 16
#define N_RELAX 100

typedef __attribute__((ext_vector_type(2))) float v2f;
typedef __attribute__((ext_vector_type(8))) float v8f;

__device__ __forceinline__ v8f wmma4(v2f a, v2f b, v8f c) {
  // D = A(16x4 f32) * B(4x16 f32) + C(16x16 f32)
  return __builtin_amdgcn_wmma_f32_16x16x4_f32(false, a, false, b, (short)0, c,
                                               false, false);
}

// ---------------------------------------------------------------------------
// Complex 128x128 GEMM on the WMMA f32 pipe.
//   Dre = s * (Are*Bre - ca*cb * Aim*Bim)
//   Dim = s * (ca*Aim*Bre + cb*Are*Bim)        (if HAS_DIM)
// ca/cb = +-1 implement conjugation of either operand. HAS_BIM=false means B
// is real (Bim ignored). strideX = 0 lets the DFT matrix W be shared across
// the batched z dimension (z = batch*3 component planes).
// ---------------------------------------------------------------------------
template <bool HAS_BIM, bool HAS_DIM>
__global__ void cgemm128(const float* __restrict__ Are, const float* __restrict__ Aim,
                         const float* __restrict__ Bre, const float* __restrict__ Bim,
                         float* __restrict__ Dre, float* __restrict__ Dim,
                         int strideA, int strideB, int strideD,
                         float ca, float cb, float cmul,
                         const float* __restrict__ scalePtr) {
  const int z = blockIdx.z;
  const float* are = Are + (size_t)z * strideA;
  const float* aim = Aim + (size_t)z * strideA;
  const float* bre = Bre + (size_t)z * strideB;
  const float* bim = HAS_BIM ? (Bim + (size_t)z * strideB) : nullptr;
  float* dre = Dre + (size_t)z * strideD;
  float* dim_ = HAS_DIM ? (Dim + (size_t)z * strideD) : nullptr;

  const int lane = threadIdx.x;                          // 0..31 (wave32)
  const int tx = blockIdx.x * blockDim.y + threadIdx.y;  // tile col 0..7
  const int ty = blockIdx.y;                             // tile row 0..7

  const int mrow = ty * 16 + (lane & 15);  // A row (per ISA 16x4 layout)
  const int ncol = tx * 16 + (lane & 15);  // B/D column
  const int koff = (lane >> 4) * 2;        // lanes 16..31 hold K+2, K+3

  const float sAB = -ca * cb;

  v8f accRe = {0.f, 0.f, 0.f, 0.f, 0.f, 0.f, 0.f, 0.f};
  v8f accIm = {0.f, 0.f, 0.f, 0.f, 0.f, 0.f, 0.f, 0.f};

  for (int k0 = 0; k0 < N; k0 += 4) {
    const int ka = k0 + koff;
    v2f a_re;
    a_re.x = are[mrow * N + ka];
    a_re.y = are[mrow * N + ka + 1];
    v2f b_re;
    b_re.x = bre[ka * N + ncol];
    b_re.y = bre[(ka + 1) * N + ncol];
    if (k0 + 4 < N) __builtin_prefetch(&bre[(ka + 4) * N + ncol], 0, 0);

    accRe = wmma4(a_re, b_re, accRe);

    v2f a_im;
    a_im.x = aim[mrow * N + ka];
    a_im.y = aim[mrow * N + ka + 1];

    if constexpr (HAS_BIM) {
      v2f b_im;
      b_im.x = bim[ka * N + ncol];
      b_im.y = bim[(ka + 1) * N + ncol];
      v2f a_im_s;
      a_im_s.x = a_im.x * sAB;
      a_im_s.y = a_im.y * sAB;
      accRe = wmma4(a_im_s, b_im, accRe);
      if constexpr (HAS_DIM) {
        v2f b_im_s;
        b_im_s.x = b_im.x * cb;
        b_im_s.y = b_im.y * cb;
        accIm = wmma4(a_re, b_im_s, accIm);
      }
    }
    if constexpr (HAS_DIM) {
      v2f a_im_c;
      a_im_c.x = a_im.x * ca;
      a_im_c.y = a_im.y * ca;
      accIm = wmma4(a_im_c, b_re, accIm);
    }
  }

  const float s = cmul * (scalePtr ? scalePtr[0] : 1.0f);
  const int half = (lane >> 4) * 8;  // C/D layout: lanes 16..31 hold M+8
  for (int r = 0; r < 8; ++r) {
    const int row = ty * 16 + half + r;
    dre[row * N + ncol] = s * accRe[r];
    if constexpr (HAS_DIM) dim_[row * N + ncol] = s * accIm[r];
  }
}

// G[i] = sum_j Kf[i,j] * F[j]  (complex, per frequency-pixel; in-place on F)
__global__ void freqmul(float* __restrict__ Fre, float* __restrict__ Fim,
                        const float* __restrict__ Kfre,
                        const float* __restrict__ Kfim, int nb) {
  const int idx = blockIdx.x * blockDim.x + threadIdx.x;
  if (idx >= nb * NN) return;
  const int p = idx % NN;
  const int b = idx / NN;
  float fr[3], fi[3];
  for (int j = 0; j < 3; ++j) {
    fr[j] = Fre[(size_t)(b * 3 + j) * NN + p];
    fi[j] = Fim[(size_t)(b * 3 + j) * NN + p];
  }
  for (int i = 0; i < 3; ++i) {
    float gr = 0.f, gi = 0.f;
    for (int j = 0; j < 3; ++j) {
      const float kr = Kfre[(size_t)(i * 3 + j) * NN + p];
      const float ki = Kfim[(size_t)(i * 3 + j) * NN + p];
      gr += kr * fr[j] - ki * fi[j];
      gi += kr * fi[j] + ki * fr[j];
    }
    Fre[(size_t)(b * 3 + i) * NN + p] = gr;
    Fim[(size_t)(b * 3 + i) * NN + p] = gi;
  }
}

// Fused: exchange Laplacian + Beff + LLG + SOT torque + RK4 stage combine.
__global__ void torque_rk4(const float* __restrict__ ms,
                           const float* __restrict__ mbase,
                           float* __restrict__ kacc, float* __restrict__ mout,
                           const float* __restrict__ bd,
                           const float* __restrict__ bext0,
                           const float* __restrict__ smask,
                           const float* __restrict__ sig, int t, int Tn,
                           const float* __restrict__ alphaMap, float alphaConst,
                           const float* __restrict__ MsatP,
                           const float* __restrict__ AexP,
                           const float* __restrict__ cSotP, float wk, float cs,
                           int firstStage, int finalStage, int nb) {
  const int idx = blockIdx.x * blockDim.x + threadIdx.x;
  if (idx >= nb * NN) return;
  const int p = idx % NN;
  const int b = idx / NN;
  const int x = p / N, y = p % N;
  const int xm = x > 0 ? x - 1 : 0, xp = x < N - 1 ? x + 1 : N - 1;
  const int ym = y > 0 ? y - 1 : 0, yp = y < N - 1 ? y + 1 : N - 1;
  const float invD2 = 1.0f / (5e-8f * 5e-8f);  // 1/DX^2 == 1/DY^2

  const float Msat = MsatP[0];
  const float exC = 2.0f * AexP[0] / Msat;
  const float cS = cSotP[0];
  const float sv = sig ? sig[b * Tn + t] : 0.0f;
  const float a = alphaMap ? alphaMap[p] : alphaConst;

  float m[3], B[3];
  for (int c = 0; c < 3; ++c) {
    const float* mp = ms + (size_t)(b * 3 + c) * NN;
    const float mc = mp[p];
    const float lap = (mp[xp * N + y] + mp[xm * N + y] - 2.0f * mc) * invD2 +
                      (mp[x * N + yp] + mp[x * N + ym] - 2.0f * mc) * invD2;
    float be = bext0[(size_t)c * NN + p];
    if (smask) be += sv * smask[(size_t)c * NN + p];
    B[c] = be + exC * lap + bd[(size_t)(b * 3 + c) * NN + p];
    m[c] = mc;
  }
  // mxB
  const float x0 = m[1] * B[2] - m[2] * B[1];
  const float x1 = m[2] * B[0] - m[0] * B[2];
  const float x2 = m[0] * B[1] - m[1] * B[0];
  // m x (mxB)
  const float y0 = m[1] * x2 - m[2] * x1;
  const float y1 = m[2] * x0 - m[0] * x2;
  const float y2 = m[0] * x1 - m[1] * x0;
  const float inv = 1.0f / (1.0f + a * a);
  float kv[3];
  kv[0] = -(x0 + a * y0) * inv;
  kv[1] = -(x1 + a * y1) * inv;
  kv[2] = -(x2 + a * y2) * inv;
  // SOT: c * m x (m x yhat); m x yhat = (-m2, 0, m0)
  const float c0 = -m[2], c1 = 0.0f, c2 = m[0];
  kv[0] += cS * (m[1] * c2 - m[2] * c1);
  kv[1] += cS * (m[2] * c0 - m[0] * c2);
  kv[2] += cS * (m[0] * c1 - m[1] * c0);

  for (int c = 0; c < 3; ++c) {
    const size_t o = (size_t)(b * 3 + c) * NN + p;
    const float ka = firstStage ? wk * kv[c] : kacc[o] + wk * kv[c];
    kacc[o] = ka;
    const float upd = finalStage ? ka : kv[c];
    mout[o] = mbase[o] + cs * upd;
  }
}

__global__ void init_w(float* __restrict__ Wre, float* __restrict__ Wim) {
  const int idx = blockIdx.x * blockDim.x + threadIdx.x;
  if (idx >= NN) return;
  const int j = idx / N, k = idx % N;
  const int prod = (j * k) & (N - 1);  // exact modular phase
  const float theta = (float)prod * (6.28318530717958647692f / (float)N);
  float s, c;
  __sincosf(theta, &s, &c);
  Wre[idx] = c;
  Wim[idx] = -s;  // exp(-2*pi*i*j*k/N)
}

__global__ void init_m0(float* __restrict__ m0) {
  const int idx = blockIdx.x * blockDim.x + threadIdx.x;
  if (idx >= 3 * NN) return;
  m0[idx] = (idx / NN == 1) ? 1.0f : 0.0f;
}

__global__ void bcast_m(float* __restrict__ m, const float* __restrict__ m0) {
  const int idx = blockIdx.x * blockDim.x + threadIdx.x;
  if (idx >= NB * 3 * NN) return;
  const int c = (idx / NN) % 3;
  m[idx] = m0[(size_t)c * NN + (idx % NN)];
}

__global__ void zero_out(float* __restrict__ out) {
  if (threadIdx.x < NB * 4) out[threadIdx.x] = 0.0f;
}

__global__ void measure(const float* __restrict__ m, const float* __restrict__ m0,
                        const int* __restrict__ probe,
                        const float* __restrict__ MsatP, float* __restrict__ out) {
  const int tid = threadIdx.x;
  if (tid >= NB * 4) return;
  const int b = tid / 4, pi = tid % 4;
  const int px = probe[pi * 2 + 0], py = probe[pi * 2 + 1];
  const int off = px * N + py;
  const float v =
      (m[((size_t)(b * 3 + 2)) * NN + off] - m0[(size_t)2 * NN + off]) * MsatP[0];
  out[tid] += v * v;
}

extern "C" void kernel_launch(void* const* d_in, const int* in_sizes, int n_in,
                              void* d_out, int out_size, void* d_ws,
                              size_t ws_size, hipStream_t stream) {
  (void)in_sizes; (void)n_in; (void)out_size; (void)ws_size;

  const float* signal   = (const float*)d_in[0];  // (16,16,1)
  const float* bext0    = (const float*)d_in[1];  // (3,128,128)
  const float* MsatP    = (const float*)d_in[2];  // scalar
  const float* alphaMap = (const float*)d_in[3];  // (1,1,128,128)
  const float* Kin      = (const float*)d_in[4];  // (3,3,128,128)
  const float* smask    = (const float*)d_in[5];  // (3,128,128)
  const float* AexP     = (const float*)d_in[6];  // scalar
  const float* cSotP    = (const float*)d_in[7];  // scalar
  const int*   probe    = (const int*)d_in[8];    // (4,2)
  float* out = (float*)d_out;

  // Workspace carve (~26.7 MB, fits well inside L2).
  float* w = (float*)d_ws;
  auto carve = [&](size_t n) { float* p = w; w += n; return p; };
  float* Wre  = carve(NN);
  float* Wim  = carve(NN);
  float* Kfre = carve(9 * NN);
  float* Kfim = carve(9 * NN);
  float* m0   = carve(3 * NN);
  float* m    = carve((size_t)NB * 3 * NN);
  float* tmpA = carve((size_t)NB * 3 * NN);
  float* tmpB = carve((size_t)NB * 3 * NN);
  float* kacc = carve((size_t)NB * 3 * NN);
  float* t1re = carve((size_t)NB * 3 * NN);
  float* t1im = carve((size_t)NB * 3 * NN);
  float* fre  = carve((size_t)NB * 3 * NN);
  float* fim  = carve((size_t)NB * 3 * NN);
  float* bd = fre;  // irow result reuses fre (dead after icol each stage)

  const dim3 gblk(32, 4, 1);

  // demag(m_planes) -> bd, Msat-scaled.  F = W*M*W, G = Kf.F, B = Re(W~ G W~)/N^2
  auto demag = [&](const float* mpl, int nb) {
    dim3 grd(2, 8, nb * 3);
    // row-forward: T1 = W * M (M real)
    cgemm128<false, true><<<grd, gblk, 0, stream>>>(
        Wre, Wim, mpl, nullptr, t1re, t1im, 0, NN, NN, 1.f, 1.f, 1.f, nullptr);
    // col-forward: F = T1 * W
    cgemm128<true, true><<<grd, gblk, 0, stream>>>(
        t1re, t1im, Wre, Wim, fre, fim, NN, 0, NN, 1.f, 1.f, 1.f, nullptr);
    freqmul<<<(nb * NN + 255) / 256, 256, 0, stream>>>(fre, fim, Kfre, Kfim, nb);
    // col-inverse: T2 = G * conj(W)
    cgemm128<true, true><<<grd, gblk, 0, stream>>>(
        fre, fim, Wre, Wim, t1re, t1im, NN, 0, NN, 1.f, -1.f, 1.f, nullptr);
    // row-inverse, real part, scaled by Msat/N^2
    cgemm128<true, false><<<grd, gblk, 0, stream>>>(
        Wre, Wim, t1re, t1im, bd, nullptr, 0, NN, NN, -1.f, 1.f,
        1.0f / (float)NN, MsatP);
  };

  const float h = 0.17595f;  // GAMMA_LL * DT
  auto rk4 = [&](float* mstate, int nb, const float* sig, int t,
                 const float* amap, float aconst, const float* sm) {
    const float* ins[4] = {mstate, tmpA, tmpB, tmpA};
    float* outs[4] = {tmpA, tmpB, tmpA, mstate};
    const float wks[4] = {1.f, 2.f, 2.f, 1.f};
    const float css[4] = {h * 0.5f, h * 0.5f, h, h / 6.f};
    for (int s = 0; s < 4; ++s) {
      demag(ins[s], nb);
      const int nthr = nb * NN;
      torque_rk4<<<(nthr + 255) / 256, 256, 0, stream>>>(
          ins[s], mstate, kacc, outs[s], bd, bext0, sm, sig, t, T_STEPS, amap,
          aconst, MsatP, AexP, cSotP, wks[s], css[s], s == 0 ? 1 : 0,
          s == 3 ? 1 : 0, nb);
    }
  };

  // ---- setup: DFT matrix, demag kernel spectrum, initial state, output ----
  init_w<<<(NN + 255) / 256, 256, 0, stream>>>(Wre, Wim);
  zero_out<<<1, 64, 0, stream>>>(out);
  {
    dim3 grd(2, 8, 9);  // Kf = fft2(K), 9 real planes
    cgemm128<false, true><<<grd, gblk, 0, stream>>>(
        Wre, Wim, Kin, nullptr, t1re, t1im, 0, NN, NN, 1.f, 1.f, 1.f, nullptr);
    cgemm128<true, true><<<grd, gblk, 0, stream>>>(
        t1re, t1im, Wre, Wim, Kfre, Kfim, NN, 0, NN, 1.f, 1.f, 1.f, nullptr);
  }
  init_m0<<<(3 * NN + 255) / 256, 256, 0, stream>>>(m0);

  // ---- relaxation: batch-invariant, so run a single batch (16x less work) ----
  for (int i = 0; i < N_RELAX; ++i)
    rk4(m0, 1, nullptr, 0, nullptr, 0.5f, nullptr);

  // ---- broadcast relaxed state, then the driven scan over 16 batches ----
  bcast_m<<<((size_t)NB * 3 * NN + 255) / 256, 256, 0, stream>>>(m, m0);
  for (int t = 0; t < T_STEPS; ++t) {
    rk4(m, NB, signal, t, alphaMap, 0.f, smask);
    measure<<<1, 64, 0, stream>>>(m, m0, probe, MsatP, out);
  }
}